// GCN_65463891526201
// MI455X (gfx1250) — compile-verified
//
#include <hip/hip_runtime.h>
#include <cstddef>

typedef __attribute__((ext_vector_type(2))) float v2f;
typedef __attribute__((ext_vector_type(8))) float v8f;

namespace {
constexpr int kNodes  = 50000;
constexpr int kEdges  = 800000;
constexpr int kGraphs = 64;
constexpr int F_IN = 128, F_H1 = 96, F_H2 = 48, F_OUT = 32;
}

// ---------------- degree / dinv / norms ----------------
__global__ void k_fill1(float* __restrict__ p, int n) {
  int i = blockIdx.x * blockDim.x + threadIdx.x;
  if (i < n) p[i] = 1.0f;  // self-loop contributes 1 to every node's degree
}

__global__ void k_zero(float* __restrict__ p, int n) {
  int i = blockIdx.x * blockDim.x + threadIdx.x;
  if (i < n) p[i] = 0.0f;
}

__global__ void k_deg_count(const int* __restrict__ dst, float* __restrict__ deg, int e) {
  int i = blockIdx.x * blockDim.x + threadIdx.x;
  if (i < e) atomicAdd(&deg[dst[i]], 1.0f);
}

// dinv = rsqrt(deg); dinv2 = 1/deg  (self-loop norm)
__global__ void k_dinv(float* __restrict__ dinv, float* __restrict__ dinv2, int n) {
  int i = blockIdx.x * blockDim.x + threadIdx.x;
  if (i < n) {
    float di = rsqrtf(dinv[i]);  // dinv currently holds degree
    dinv[i]  = di;
    dinv2[i] = di * di;
  }
}

// norm[e] = dinv[src[e]] * dinv[dst[e]]  (computed once, reused by all layers)
__global__ void k_edge_norm(const float* __restrict__ dinv, const int* __restrict__ src,
                            const int* __restrict__ dst, float* __restrict__ norm, int e) {
  int i = blockIdx.x * blockDim.x + threadIdx.x;
  if (i < e) norm[i] = dinv[src[i]] * dinv[dst[i]];
}

// ---------------- WMMA GEMM, fused epilogue ----------------
// One wave computes a full 16-row strip across NT 16-col tiles (NO = NT*16),
// so X is read exactly once. A (16x4 fp32 / step): lane m = lane&15;
// koff = (lane>>4)*2; VGPR0=K=koff, VGPR1=K=koff+1. B mirrors A with n = lane&15.
// C/D: VGPR j -> D[(lane>>4)*8 + j][lane&15].
// Epilogue writes raw H (for the edge gather) AND OUT = bias + H*dinv2
// (self-loop + bias), eliminating the separate agg_init pass.
template <int NT, bool RELU_IN>
__global__ __launch_bounds__(32) void k_gemm_wmma_fused(
    const float* __restrict__ X, const float* __restrict__ W,
    const float* __restrict__ bias, const float* __restrict__ dinv2,
    float* __restrict__ H, float* __restrict__ OUT, int K) {
  constexpr int NO = NT * 16;
  const int mtile = blockIdx.x * 16;
  const int lane  = threadIdx.x;
  const int i15   = lane & 15;
  const int koff  = (lane >> 4) << 1;

  const float* xrow = X + (size_t)(mtile + i15) * K;

  v8f acc[NT];
#pragma unroll
  for (int t = 0; t < NT; ++t) acc[t] = {};

  for (int k = 0; k < K; k += 4) {
    float a0 = xrow[k + koff];
    float a1 = xrow[k + koff + 1];
    if (RELU_IN) { a0 = fmaxf(a0, 0.0f); a1 = fmaxf(a1, 0.0f); }
    v2f a; a.x = a0; a.y = a1;
    const float* wrow0 = W + (size_t)(k + koff) * NO + i15;
    const float* wrow1 = W + (size_t)(k + koff + 1) * NO + i15;
#pragma unroll
    for (int t = 0; t < NT; ++t) {
      v2f b; b.x = wrow0[t * 16]; b.y = wrow1[t * 16];
      acc[t] = __builtin_amdgcn_wmma_f32_16x16x4_f32(
          /*neg_a=*/false, a, /*neg_b=*/false, b,
          /*c_mod=*/(short)0, acc[t], /*reuse_a=*/false, /*reuse_b=*/false);
    }
  }

  const int mbase = (lane >> 4) * 8;
#pragma unroll
  for (int j = 0; j < 8; ++j) {
    const int m = mtile + mbase + j;
    const float d2 = dinv2[m];
    float* hrow = H   + (size_t)m * NO + i15;
    float* orow = OUT + (size_t)m * NO + i15;
#pragma unroll
    for (int t = 0; t < NT; ++t) {
      float h = acc[t][j];
      hrow[t * 16] = h;
      orow[t * 16] = bias[t * 16 + i15] + h * d2;
    }
  }
}

// ---------------- edge scatter ----------------
// For each edge e, each thread handles 4 features:
// out[dst][4q..4q+3] += h[src][4q..4q+3] * norm[e]
__global__ void k_agg_edges(const float* __restrict__ h, const float* __restrict__ norm,
                            const int* __restrict__ src, const int* __restrict__ dst,
                            float* __restrict__ out, int e, int fq, int f) {
  int idx = blockIdx.x * blockDim.x + threadIdx.x;
  if (idx >= e * fq) return;
  int ed = idx / fq;
  int q  = idx - ed * fq;
  int s = src[ed];
  int d = dst[ed];
  float nrm = norm[ed];
  const float4* hp = reinterpret_cast<const float4*>(h + (size_t)s * f);
  float4 v = hp[q];
  float* op = out + (size_t)d * f + q * 4;
  atomicAdd(op + 0, v.x * nrm);
  atomicAdd(op + 1, v.y * nrm);
  atomicAdd(op + 2, v.z * nrm);
  atomicAdd(op + 3, v.w * nrm);
}

// ---------------- global mean pool (final ReLU fused on read) ----------------
__global__ void k_pool_accum(const float* __restrict__ h, const int* __restrict__ batch,
                             float* __restrict__ sums, int n, int f) {
  int idx = blockIdx.x * blockDim.x + threadIdx.x;
  if (idx >= n * f) return;
  int i = idx / f;
  int c = idx - i * f;
  float v = fmaxf(h[idx], 0.0f);
  atomicAdd(&sums[(size_t)batch[i] * f + c], v);
}

__global__ void k_pool_count(const int* __restrict__ batch, float* __restrict__ cnt, int n) {
  int i = blockIdx.x * blockDim.x + threadIdx.x;
  if (i < n) atomicAdd(&cnt[batch[i]], 1.0f);
}

__global__ void k_pool_final(const float* __restrict__ sums, const float* __restrict__ cnt,
                             float* __restrict__ out, int g, int f) {
  int idx = blockIdx.x * blockDim.x + threadIdx.x;
  if (idx >= g * f) return;
  out[idx] = sums[idx] / fmaxf(cnt[idx / f], 1.0f);
}

// ---------------- launcher ----------------
extern "C" void kernel_launch(void* const* d_in, const int* in_sizes, int n_in,
                              void* d_out, int out_size, void* d_ws, size_t ws_size,
                              hipStream_t stream) {
  (void)in_sizes; (void)n_in; (void)out_size; (void)ws_size;
  const float* x     = (const float*)d_in[0];
  const int*   ei    = (const int*)d_in[1];
  const int*   batch = (const int*)d_in[2];
  const float* W1 = (const float*)d_in[3];
  const float* b1 = (const float*)d_in[4];
  const float* W2 = (const float*)d_in[5];
  const float* b2 = (const float*)d_in[6];
  const float* W3 = (const float*)d_in[7];
  const float* b3 = (const float*)d_in[8];
  float* out = (float*)d_out;

  const int* src = ei;            // edge_index[0]
  const int* dst = ei + kEdges;   // edge_index[1]

  float* ws = (float*)d_ws;
  size_t off = 0;
  auto take = [&](size_t nfloats) {
    float* p = ws + off;
    off += (nfloats + 63) & ~(size_t)63;  // 256B-align each region
    return p;
  };
  float* dinv  = take(kNodes);
  float* dinv2 = take(kNodes);
  float* enorm = take(kEdges);
  float* bufA  = take((size_t)kNodes * F_H1);  // rotating buffers
  float* bufB  = take((size_t)kNodes * F_H1);
  float* bufC  = take((size_t)kNodes * F_H1);
  float* sums  = take((size_t)kGraphs * F_OUT);
  float* cnt   = take(kGraphs);

  const int T = 256;
  auto nb = [&](long long n) { return (unsigned)((n + T - 1) / T); };

  // degrees -> dinv, dinv2, per-edge norms
  k_fill1<<<nb(kNodes), T, 0, stream>>>(dinv, kNodes);
  k_deg_count<<<nb(kEdges), T, 0, stream>>>(dst, dinv, kEdges);
  k_dinv<<<nb(kNodes), T, 0, stream>>>(dinv, dinv2, kNodes);
  k_edge_norm<<<nb(kEdges), T, 0, stream>>>(dinv, src, dst, enorm, kEdges);

  const unsigned MB = kNodes / 16;  // 3125 m-tiles, one wave each

  // ---- Layer 1: h1 = x @ W1 -> bufA ; a1 (init b1 + h1*dinv2) -> bufB ; scatter ----
  k_gemm_wmma_fused<F_H1 / 16, false><<<MB, 32, 0, stream>>>(x, W1, b1, dinv2, bufA, bufB, F_IN);
  k_agg_edges<<<nb((long long)kEdges * (F_H1 / 4)), T, 0, stream>>>(bufA, enorm, src, dst, bufB,
                                                                    kEdges, F_H1 / 4, F_H1);

  // ---- Layer 2: h2 = relu(a1=bufB) @ W2 -> bufA ; a2 -> bufC ; scatter ----
  k_gemm_wmma_fused<F_H2 / 16, true><<<MB, 32, 0, stream>>>(bufB, W2, b2, dinv2, bufA, bufC, F_H1);
  k_agg_edges<<<nb((long long)kEdges * (F_H2 / 4)), T, 0, stream>>>(bufA, enorm, src, dst, bufC,
                                                                    kEdges, F_H2 / 4, F_H2);

  // ---- Layer 3: h3 = relu(a2=bufC) @ W3 -> bufA ; a3 -> bufB ; scatter ----
  k_gemm_wmma_fused<F_OUT / 16, true><<<MB, 32, 0, stream>>>(bufC, W3, b3, dinv2, bufA, bufB, F_H2);
  k_agg_edges<<<nb((long long)kEdges * (F_OUT / 4)), T, 0, stream>>>(bufA, enorm, src, dst, bufB,
                                                                     kEdges, F_OUT / 4, F_OUT);

  // ---- Global mean pool over graphs (relu(a3) fused into accumulation) ----
  k_zero<<<nb(kGraphs * F_OUT), T, 0, stream>>>(sums, kGraphs * F_OUT);
  k_zero<<<nb(kGraphs), T, 0, stream>>>(cnt, kGraphs);
  k_pool_accum<<<nb((long long)kNodes * F_OUT), T, 0, stream>>>(bufB, batch, sums, kNodes, F_OUT);
  k_pool_count<<<nb(kNodes), T, 0, stream>>>(batch, cnt, kNodes);
  k_pool_final<<<nb(kGraphs * F_OUT), T, 0, stream>>>(sums, cnt, out, kGraphs, F_OUT);
}